// PSROIPool_22179211117166
// MI455X (gfx1250) — compile-verified
//
#include <hip/hip_runtime.h>

typedef __attribute__((ext_vector_type(2))) float v2f;
typedef __attribute__((ext_vector_type(8))) float v8f;

#define HH 80
#define WW 80
#define LDS_STRIDE 84            // 84*4B = 336B rows: 16B aligned, bank-conflict-free gathers
#define SPATIAL_SCALE 0.0625f

// One workgroup per (b,c) plane. Builds the 80x80 inclusive SAT in LDS using
// f32 WMMA (prefix-sum == matmul with a triangular ones matrix, generated in
// registers), then does the position-sensitive ROI pooling from LDS.
__global__ __launch_bounds__(256)
void psroi_sat_wmma_kernel(const float* __restrict__ input,
                           const float* __restrict__ rois,
                           float* __restrict__ out,
                           int B, int C, int N)
{
    __shared__ float sX[HH * LDS_STRIDE];   // input plane, later the SAT (Z)
    __shared__ float sY[HH * LDS_STRIDE];   // row-cumsum intermediate

    const int plane = blockIdx.x;           // b*C + c
    const int b = plane / C;
    const int c = plane - b * C;

    const int tid  = threadIdx.x;
    // Wave id is uniform within a wave; route it through readfirstlane so the
    // compiler proves it and emits scalar (SCC-based) loop control around the
    // WMMAs instead of per-lane EXEC-masked loops. Guarantees EXEC==all-ones
    // structurally at every WMMA, as the ISA requires.
    const int wv   = __builtin_amdgcn_readfirstlane(tid >> 5);  // 0..7
    const int lane = tid & 31;
    const int half = lane >> 4;             // 0 or 1 (lane half, per WMMA layouts)
    const int ln   = lane & 15;
    const int k_lo = 2 * half;              // this lane's K sub-offset for A/B operands

    // ---- stage plane into LDS (float4, rows never split since 80 % 4 == 0) ----
    const float* src = input + (size_t)plane * (HH * WW);
    for (int i4 = tid; i4 < (HH * WW) / 4; i4 += 256) {
        int flat = i4 * 4;
        int r    = flat / WW;
        int col  = flat - r * WW;
        float4 v = ((const float4*)src)[i4];
        *(float4*)&sX[r * LDS_STRIDE + col] = v;
    }
    __syncthreads();

    // ---- pass 1: sY = X * U  (row-inclusive cumsum), U[k][n] = (k <= n) ----
    // 25 output tiles (5 row-blocks x 5 col-tiles), wave-uniform distribution.
    for (int t = wv; t < 25; t += 8) {
        const int mb = t / 5;
        const int jt = t - mb * 5;
        const int m_base = mb * 16;
        const int n_base = jt * 16;
        const int n_g = n_base + ln;        // this lane's output column
        v8f acc = {};
        for (int kc = 0; kc <= jt; ++kc) {
            #pragma unroll
            for (int q = 0; q < 4; ++q) {
                const int k0 = kc * 16 + q * 4;
                const int kg = k0 + k_lo;
                // A = X chunk (16x4): lanes 0-15 -> K = k0,k0+1 ; lanes 16-31 -> K = k0+2,k0+3
                const int arow = m_base + ln;
                v2f a;
                a.x = sX[arow * LDS_STRIDE + kg];
                a.y = sX[arow * LDS_STRIDE + kg + 1];
                // B = U chunk (4x16), generated in registers
                v2f bm;
                bm.x = (kg     <= n_g) ? 1.0f : 0.0f;
                bm.y = (kg + 1 <= n_g) ? 1.0f : 0.0f;
                acc = __builtin_amdgcn_wmma_f32_16x16x4_f32(
                          false, a, false, bm, (short)0, acc, false, false);
            }
        }
        #pragma unroll
        for (int r = 0; r < 8; ++r)
            sY[(m_base + r + 8 * half) * LDS_STRIDE + (n_base + ln)] = acc[r];
    }
    __syncthreads();

    // ---- pass 2: sX = L * Y  (col-inclusive cumsum), L[m][k] = (k <= m) ----
    for (int t = wv; t < 25; t += 8) {
        const int mb = t / 5;
        const int jt = t - mb * 5;
        const int m_base = mb * 16;
        const int n_base = jt * 16;
        const int m_g = m_base + ln;
        const int col = n_base + ln;
        v8f acc = {};
        for (int kc = 0; kc <= mb; ++kc) {
            #pragma unroll
            for (int q = 0; q < 4; ++q) {
                const int k0 = kc * 16 + q * 4;
                const int kg = k0 + k_lo;
                // A = L chunk (16x4), generated in registers
                v2f a;
                a.x = (kg     <= m_g) ? 1.0f : 0.0f;
                a.y = (kg + 1 <= m_g) ? 1.0f : 0.0f;
                // B = Y chunk (4x16): VGPR v holds row kg + v, this lane's column
                v2f bm;
                bm.x = sY[kg       * LDS_STRIDE + col];
                bm.y = sY[(kg + 1) * LDS_STRIDE + col];
                acc = __builtin_amdgcn_wmma_f32_16x16x4_f32(
                          false, a, false, bm, (short)0, acc, false, false);
            }
        }
        #pragma unroll
        for (int r = 0; r < 8; ++r)
            sX[(m_base + r + 8 * half) * LDS_STRIDE + (n_base + ln)] = acc[r];
    }
    __syncthreads();

    // ---- pooling: sX now holds Z with Z[i][j] = SAT[i+1][j+1] ----
    const int d   = c / 49;
    const int rem = c - d * 49;
    const int ph  = rem / 7;
    const int pw  = rem - ph * 7;
    const float phf = (float)ph, pwf = (float)pw;
    const int D = C / 49;

    for (int n = tid; n < N; n += 256) {
        const float* r5 = rois + (size_t)n * 5;
        if ((int)r5[0] != b) continue;
        // match reference rounding exactly: rnd(v) = floor(v + 0.5)
        float roi_sw = floorf(r5[1] + 0.5f) * SPATIAL_SCALE;
        float roi_sh = floorf(r5[2] + 0.5f) * SPATIAL_SCALE;
        float roi_ew = floorf((r5[3] + 1.0f) + 0.5f) * SPATIAL_SCALE;
        float roi_eh = floorf((r5[4] + 1.0f) + 0.5f) * SPATIAL_SCALE;
        float roi_w = fmaxf(roi_ew - roi_sw, 0.1f);
        float roi_h = fmaxf(roi_eh - roi_sh, 0.1f);
        float bin_h = roi_h * (1.0f / 7.0f);
        float bin_w = roi_w * (1.0f / 7.0f);
        int hs = (int)fminf(fmaxf(floorf(phf * bin_h + roi_sh), 0.0f), 80.0f);
        int he = (int)fminf(fmaxf(ceilf((phf + 1.0f) * bin_h + roi_sh), 0.0f), 80.0f);
        int ws = (int)fminf(fmaxf(floorf(pwf * bin_w + roi_sw), 0.0f), 80.0f);
        int we = (int)fminf(fmaxf(ceilf((pwf + 1.0f) * bin_w + roi_sw), 0.0f), 80.0f);

        float s_hewe = (he > 0 && we > 0) ? sX[(he - 1) * LDS_STRIDE + (we - 1)] : 0.0f;
        float s_hswe = (hs > 0 && we > 0) ? sX[(hs - 1) * LDS_STRIDE + (we - 1)] : 0.0f;
        float s_hews = (he > 0 && ws > 0) ? sX[(he - 1) * LDS_STRIDE + (ws - 1)] : 0.0f;
        float s_hsws = (hs > 0 && ws > 0) ? sX[(hs - 1) * LDS_STRIDE + (ws - 1)] : 0.0f;
        float s = s_hewe - s_hswe - s_hews + s_hsws;

        int area = (he - hs) * (we - ws);
        float val = (area > 0) ? s / (float)area : 0.0f;
        out[((size_t)n * D + d) * 49 + ph * 7 + pw] = val;
    }
}

extern "C" void kernel_launch(void* const* d_in, const int* in_sizes, int n_in,
                              void* d_out, int out_size, void* d_ws, size_t ws_size,
                              hipStream_t stream) {
    const float* input = (const float*)d_in[0];
    const float* rois  = (const float*)d_in[1];
    float* out = (float*)d_out;

    const int C = 1029;
    const int planeElems = HH * WW;
    int B = in_sizes[0] / (C * planeElems);   // 4
    int N = in_sizes[1] / 5;                  // 1024

    dim3 grid(B * C);                         // one workgroup per (b,c) plane
    psroi_sat_wmma_kernel<<<grid, 256, 0, stream>>>(input, rois, out, B, C, N);
}